// SparseGraphConvolutionLayer_36532991820137
// MI455X (gfx1250) — compile-verified
//
#include <hip/hip_runtime.h>

typedef __attribute__((ext_vector_type(2))) float v2f;
typedef __attribute__((ext_vector_type(8))) float v8f;

#define IN_F   256
#define OUT_F  128
#define LDS_STRIDE 260   // 256 + 4 dwords pad: row stride = 4 banks -> conflict-free column reads

// ---------------------------------------------------------------------------
// Kernel 1: h[N,128] = x[N,256] @ w[256,128] using V_WMMA_F32_16X16X4_F32.
// Block = 256 threads = 8 waves. Block b computes M-tile b (rows 16b..16b+15);
// wave w computes N-tile w (cols 16w..16w+15). K swept in steps of 4.
// ---------------------------------------------------------------------------
__global__ __launch_bounds__(256) void sgc_gemm_wmma_f32(
    const float* __restrict__ x, const float* __restrict__ w,
    float* __restrict__ h)
{
    __shared__ float ldsA[16 * LDS_STRIDE];   // 16 rows x 256 cols of x, padded

    const int tid  = threadIdx.x;
    const int lane = tid & 31;
    const int wave = tid >> 5;
    const int m0   = blockIdx.x * 16;

    // Cooperative load of the 16x256 x-tile into LDS: 1024 float4s / 256 threads.
    {
        const float4* xg = (const float4*)(x + (size_t)m0 * IN_F);
        #pragma unroll
        for (int i = 0; i < 4; ++i) {
            int f4   = tid + i * 256;         // 0..1023
            int row  = f4 >> 6;               // 64 float4 per row
            int col4 = f4 & 63;
            float4 v = xg[(size_t)row * 64 + col4];
            *(float4*)&ldsA[row * LDS_STRIDE + col4 * 4] = v;
        }
    }
    __syncthreads();

    const int laneM = lane & 15;
    const int hi    = lane >> 4;              // 0: lanes 0-15, 1: lanes 16-31
    const int n0    = wave * 16;

    // A per-lane: row M=laneM, K-pair {2*hi, 2*hi+1} within each K-step.
    const float* arow = &ldsA[laneM * LDS_STRIDE];
    // B per-lane: column N = n0+laneM, rows K (stride OUT_F).
    const float* wcol = w + n0 + laneM;

    v8f acc = {};
    #pragma unroll 8
    for (int k = 0; k < IN_F; k += 4) {
        const int ka = k + 2 * hi;
        v2f a;
        a.x = arow[ka];
        a.y = arow[ka + 1];
        v2f b;
        b.x = wcol[ka * OUT_F];
        b.y = wcol[(ka + 1) * OUT_F];
        // 8 args: (neg_a, A, neg_b, B, c_mod, C, reuse_a, reuse_b)
        acc = __builtin_amdgcn_wmma_f32_16x16x4_f32(
            false, a, false, b, (short)0, acc, false, false);
    }

    // C/D layout: VGPR v holds M = v + 8*hi, N = laneM.
    float* hout = h + (size_t)(m0 + 8 * hi) * OUT_F + n0 + laneM;
    #pragma unroll
    for (int v = 0; v < 8; ++v) {
        hout[(size_t)v * OUT_F] = acc[v];
    }
}

// ---------------------------------------------------------------------------
// Kernel 2: zero-init the output (harness poisons d_out).
// ---------------------------------------------------------------------------
__global__ __launch_bounds__(256) void sgc_zero(float* __restrict__ out, int n)
{
    int i = blockIdx.x * 256 + threadIdx.x;
    if (i < n) out[i] = 0.0f;
}

// ---------------------------------------------------------------------------
// Kernel 3: out[rows[e]] += h[cols[e]] for all edges.
// One wave per edge: lane l handles features 4l..4l+3 (b128 gather from
// L2-resident h, 4x global_atomic_add_f32 into out).
// ---------------------------------------------------------------------------
__global__ __launch_bounds__(256) void sgc_scatter_add(
    const float* __restrict__ h,
    const int* __restrict__ rows, const int* __restrict__ cols,
    float* __restrict__ out, int n_edges)
{
    const int lane = threadIdx.x & 31;
    const int edge = (blockIdx.x * 256 + (int)threadIdx.x) >> 5;
    if (edge >= n_edges) return;

    const int r = rows[edge];
    const int c = cols[edge];

    const float4 v = *(const float4*)(h + (size_t)c * OUT_F + lane * 4);
    float* o = out + (size_t)r * OUT_F + lane * 4;
    atomicAdd(o + 0, v.x);
    atomicAdd(o + 1, v.y);
    atomicAdd(o + 2, v.z);
    atomicAdd(o + 3, v.w);
}

// ---------------------------------------------------------------------------
extern "C" void kernel_launch(void* const* d_in, const int* in_sizes, int n_in,
                              void* d_out, int out_size, void* d_ws, size_t ws_size,
                              hipStream_t stream) {
    const float* x     = (const float*)d_in[0];   // [N, 256] f32
    const float* w     = (const float*)d_in[1];   // [256, 128] f32
    const int*   erows = (const int*)  d_in[2];   // [E] i32
    const int*   ecols = (const int*)  d_in[3];   // [E] i32
    float*       out   = (float*)d_out;           // [N, 128] f32
    float*       h     = (float*)d_ws;            // scratch: [N, 128] f32 (25.6 MB)

    const int n_nodes = in_sizes[0] / IN_F;       // 50000
    const int n_edges = in_sizes[2];              // 1,600,000

    // Phase 1: dense projection via f32 WMMA (50000 % 16 == 0 -> exact tiling).
    sgc_gemm_wmma_f32<<<n_nodes / 16, 256, 0, stream>>>(x, w, h);

    // Phase 2a: zero the output.
    const int out_elems = n_nodes * OUT_F;
    sgc_zero<<<(out_elems + 255) / 256, 256, 0, stream>>>(out, out_elems);

    // Phase 2b: edge scatter-add (8 edges per 256-thread block).
    sgc_scatter_add<<<(n_edges + 7) / 8, 256, 0, stream>>>(h, erows, ecols, out, n_edges);
}